// GCNStage4_ReduceSum_ONNX_51994874085829
// MI455X (gfx1250) — compile-verified
//
#include <hip/hip_runtime.h>

#define FEAT     256   // feature dim (fixed by the reference)
#define EPB      16    // edges staged per block (16 rows * 1 KB = 16 KB LDS)
#define THREADS  256   // 8 wave32s per block

typedef unsigned int u32x4 __attribute__((ext_vector_type(4)));
typedef int          i32x4 __attribute__((ext_vector_type(4)));
typedef int          i32x8 __attribute__((ext_vector_type(8)));

// ---------------------------------------------------------------------------
// Pass 1: zero the accumulator (harness poisons d_out with 0xAA).
// ---------------------------------------------------------------------------
__global__ void GCN_zero_kernel(float4* __restrict__ out, int n4) {
    int i = blockIdx.x * blockDim.x + threadIdx.x;
    if (i < n4) out[i] = float4{0.f, 0.f, 0.f, 0.f};
}

// ---------------------------------------------------------------------------
// Pass 2: TDM-stage a tile of msg rows into LDS (one descriptor per block,
// tracked with TENSORcnt), then scatter-add into out[node,:] with global
// f32 atomics (resolved in L2; the 10 MB output is L2-resident).
// ---------------------------------------------------------------------------
__global__ __launch_bounds__(THREADS) void GCN_scatter_add_kernel(
    const float* __restrict__ msg,
    const long long* __restrict__ tgt,   // edge_index row 1 (int64 targets)
    float* __restrict__ out,
    int num_edges)
{
    __shared__ __align__(16) float     tile[EPB * FEAT];   // 16 KB
    __shared__            long long    s_node[EPB];

    const int      e0 = blockIdx.x * EPB;
    const unsigned t  = threadIdx.x;
    const int edges_here = (num_edges - e0 < EPB) ? (num_edges - e0) : EPB;

    // Cache the tile's target indices in LDS (one global load each).
    if (t < (unsigned)EPB) {
        s_node[t] = (e0 + (int)t < num_edges) ? tgt[e0 + (int)t] : 0;
    }

    // Low 32 bits of an LDS object's flat address are its LDS byte offset
    // (ISA 10.2).
    const unsigned lds_base = (unsigned)(unsigned long long)(void*)&tile[0];

    // --- TDM: one descriptor moves edges_here x 256 f32 (up to 16 KB). ---
    // Only wave 0 issues it (TENSORcnt is per-wave); barrier publishes LDS.
    if (t < 32u) {
        const unsigned long long ga =
            (unsigned long long)(msg + (size_t)e0 * FEAT);

        // D# group 0: count=1 | lds_addr | global_addr[56:0] | type=2
        u32x4 g0;
        g0[0] = 1u;                               // count=1, user mode
        g0[1] = lds_base;                         // lds_addr (bytes)
        g0[2] = (unsigned)ga;                     // global_addr[31:0]
        g0[3] = ((unsigned)(ga >> 32) & 0x01FFFFFFu) | (2u << 30); // type=2

        // D# group 1: data_size=4B(enc 2); 2-D tensor/tile:
        //   tensor_dim0 = 256, tensor_dim1 = edges_here,
        //   tile_dim0   = 256, tile_dim1   = edges_here,
        //   tensor_dim0_stride = 256 elements. No pad/iterate/multicast.
        i32x8 g1;
        g1[0] = (int)(2u << 16);                  // data_size=2 (4 bytes)
        g1[1] = (int)(256u << 16);                // tensor_dim0[15:0]=256
        g1[2] = (int)((unsigned)edges_here << 16);// tensor_dim1[15:0]
        g1[3] = (int)(256u << 16);                // tile_dim0=256
        g1[4] = (int)((unsigned)edges_here & 0xFFFFu); // tile_dim1
        g1[5] = 256;                              // tensor_dim0_stride[31:0]
        g1[6] = 0;
        g1[7] = 0;

        i32x4 g2 = {0, 0, 0, 0};                  // unused (<=2D tensor)
        i32x4 g3 = {0, 0, 0, 0};

#if __clang_major__ >= 23
        i32x8 g4 = {0, 0, 0, 0, 0, 0, 0, 0};
        __builtin_amdgcn_tensor_load_to_lds(g0, g1, g2, g3, g4, 0);
#else
        __builtin_amdgcn_tensor_load_to_lds(g0, g1, g2, g3, 0);
#endif
        __builtin_amdgcn_s_wait_tensorcnt(0);
    }
    __syncthreads();

    // --- Scatter: EPB edges * 64 float4-quads, 4 quads per thread. ---
    const int quads = EPB * (FEAT / 4);
#pragma unroll
    for (int q = (int)t; q < quads; q += THREADS) {
        const int le = q >> 6;            // local edge 0..EPB-1
        const int e  = e0 + le;
        if (e < num_edges) {
            const int       qi   = q & 63;               // quad within row
            const long long node = s_node[le];
            const float4 v = *(const float4*)&tile[le * FEAT + qi * 4];
            float* dst = out + (size_t)node * FEAT + (size_t)(qi * 4);
            atomicAdd(dst + 0, v.x);
            atomicAdd(dst + 1, v.y);
            atomicAdd(dst + 2, v.z);
            atomicAdd(dst + 3, v.w);
        }
    }
}

// ---------------------------------------------------------------------------
extern "C" void kernel_launch(void* const* d_in, const int* in_sizes, int n_in,
                              void* d_out, int out_size, void* d_ws, size_t ws_size,
                              hipStream_t stream) {
    const float*     msg        = (const float*)d_in[0];
    const long long* edge_index = (const long long*)d_in[1];  // [2, E] int64

    const int        num_edges = in_sizes[1] / 2;
    const long long* tgt       = edge_index + num_edges;      // row 1 = targets
    float*           out       = (float*)d_out;

    // Zero the accumulator.
    const int n4 = out_size / 4;
    GCN_zero_kernel<<<(n4 + 255) / 256, 256, 0, stream>>>((float4*)out, n4);

    // TDM-staged scatter-add.
    const int blocks = (num_edges + EPB - 1) / EPB;
    GCN_scatter_add_kernel<<<blocks, THREADS, 0, stream>>>(msg, tgt, out,
                                                           num_edges);
}